// Attend_15736760173195
// MI455X (gfx1250) — compile-verified
//
#include <hip/hip_runtime.h>
#include <hip/hip_bf16.h>

// Problem constants (match reference)
constexpr int Bc = 4;
constexpr int Sc = 512;
constexpr int Ec = 512;
constexpr int Uc = 256;

typedef __attribute__((ext_vector_type(2))) float v2f;
typedef __attribute__((ext_vector_type(8))) float v8f;

// Hardware tanh: builtin if the toolchain exposes it, else guaranteed
// v_tanh_f32 via inline asm (with v_nop for the TRANS result-use hazard,
// since the compiler cannot schedule around opaque asm).
__device__ __forceinline__ float fast_tanh(float x) {
#if __has_builtin(__builtin_amdgcn_tanhf)
    return __builtin_amdgcn_tanhf(x);   // v_tanh_f32 (gfx1250 TRANS pipe)
#else
    float r;
    asm volatile("v_tanh_f32 %0, %1\n\tv_nop" : "=v"(r) : "v"(x));
    return r;
#endif
}

__device__ __forceinline__ float fast_exp2(float x) {
#if __has_builtin(__builtin_amdgcn_exp2f)
    return __builtin_amdgcn_exp2f(x);   // v_exp_f32
#else
    float r;
    asm volatile("v_exp_f32 %0, %1\n\tv_nop" : "=v"(r) : "v"(x));
    return r;
#endif
}

// ---------------------------------------------------------------------------
// Row-major fp32 GEMM: C[M,N] = A[M,K] @ B[K,N] with V_WMMA_F32_16X16X4_F32.
// N, K and batch strides are template constants so the backend folds the
// strided B accesses into global_load immediate offsets (no per-iteration
// 64-bit address math). One wave32 per 16x16 C tile; 4 waves per block.
// grid = (N/64, M/16, batch).
//
// WMMA fp32 operand layout (ISA 7.12.2):
//   A (16x4):  lane.lo(0..15)=M, VGPR j = K = 2*hi + j   -> contiguous float2
//   B (4x16):  lane.lo=N,        VGPR j = K = 2*hi + j   -> two strided dwords
//   C/D:       VGPR r -> M = r + 8*hi, N = lane.lo
// ---------------------------------------------------------------------------
template <int N, int K, long SA, long SB, long SC>
__global__ __launch_bounds__(128) void wmma_gemm_f32(
    const float* __restrict__ A, const float* __restrict__ B,
    float* __restrict__ C)
{
    const int wave = threadIdx.x >> 5;
    const int lane = threadIdx.x & 31;
    const int lo   = lane & 15;
    const int hi   = lane >> 4;

    const int tn = blockIdx.x * 4 + wave;   // N tile
    const int tm = blockIdx.y;              // M tile
    const long zb = blockIdx.z;

    A += zb * SA;
    B += zb * SB;
    C += zb * SC;

    const float* arow = A + (long)(tm * 16 + lo) * K + 2 * hi;   // K-contiguous
    const float* bp   = B + (long)(2 * hi) * N + tn * 16 + lo;   // stride N

    v8f acc = {};
#pragma unroll 8
    for (int k = 0; k < K; k += 4) {
        v2f a = *(const v2f*)(arow + k);
        v2f b;
        b[0] = bp[(long)k * N];        // constant fold -> immediate offsets
        b[1] = bp[(long)k * N + N];
        acc = __builtin_amdgcn_wmma_f32_16x16x4_f32(
            /*neg_a=*/false, a, /*neg_b=*/false, b,
            /*c_mod=*/(short)0, acc, /*reuse_a=*/false, /*reuse_b=*/false);
    }

    float* cp = C + (long)(tm * 16 + 8 * hi) * N + tn * 16 + lo;
#pragma unroll
    for (int r = 0; r < 8; ++r)
        cp[(long)r * N] = acc[r];
}

// ---------------------------------------------------------------------------
// Fused additive-attention scores + softmax. One block per (b, i) query row:
//   sc[j] = sum_u tanh(q[i,u] + k[j,u] + b1[u]) * v[u] + b2   (j = 0..S-1)
//   P[b,i,:] = softmax(sc)
// q-row (pre-biased) and v staged in LDS; each thread owns two j rows and
// streams its key rows as float4 (L2-resident, reused line within the u-loop).
// 268M v_tanh_f32 total — this kernel is the roofline; all fp32 to match
// the reference numerics.
// ---------------------------------------------------------------------------
__global__ __launch_bounds__(256) void attend_scores_softmax(
    const float* __restrict__ qry,   // (B*S, U)
    const float* __restrict__ key,   // (B, S, U)
    const float* __restrict__ b1,    // (U)
    const float* __restrict__ v,     // (U)
    const float* __restrict__ b2,    // (1)
    float* __restrict__ P)           // (B*S, S)
{
    __shared__ float qb[Uc];     // q + b1
    __shared__ float vs[Uc];
    __shared__ float sc[Sc];
    __shared__ float red[256];

    const int row = blockIdx.x;          // b*S + i
    const int b   = row >> 9;            // row / Sc
    const int t   = threadIdx.x;

    qb[t] = qry[(long)row * Uc + t] + b1[t];
    vs[t] = v[t];
    __syncthreads();

    const float bias2 = b2[0];
    const float* kb = key + (long)b * Sc * Uc;

    // Each thread: rows j = t and j = t + 256
#pragma unroll
    for (int jj = 0; jj < 2; ++jj) {
        const int j = t + jj * 256;
        const float* krow = kb + (long)j * Uc;
        float s = 0.0f;
#pragma unroll 4
        for (int u = 0; u < Uc; u += 4) {
            float4 kv = *(const float4*)(krow + u);
            s += fast_tanh(qb[u + 0] + kv.x) * vs[u + 0];
            s += fast_tanh(qb[u + 1] + kv.y) * vs[u + 1];
            s += fast_tanh(qb[u + 2] + kv.z) * vs[u + 2];
            s += fast_tanh(qb[u + 3] + kv.w) * vs[u + 3];
        }
        sc[j] = s + bias2;
    }
    __syncthreads();

    // --- softmax over sc[0..S) ---
    float m = fmaxf(sc[t], sc[t + 256]);
    red[t] = m;
    __syncthreads();
#pragma unroll
    for (int s2 = 128; s2 > 0; s2 >>= 1) {
        if (t < s2) red[t] = fmaxf(red[t], red[t + s2]);
        __syncthreads();
    }
    const float rowmax = red[0];
    __syncthreads();

    constexpr float L2E = 1.44269504088896340736f;
    const float e0 = fast_exp2((sc[t]       - rowmax) * L2E);
    const float e1 = fast_exp2((sc[t + 256] - rowmax) * L2E);

    red[t] = e0 + e1;
    __syncthreads();
#pragma unroll
    for (int s2 = 128; s2 > 0; s2 >>= 1) {
        if (t < s2) red[t] += red[t + s2];
        __syncthreads();
    }
    const float inv = 1.0f / red[0];

    P[(long)row * Sc + t]       = e0 * inv;
    P[(long)row * Sc + t + 256] = e1 * inv;
}

// ---------------------------------------------------------------------------
// Launch: key_pre = h1 @ w[:E]; qry_pre = h2 @ w[E:]; P = softmax(scores);
// out = P @ h1 (batched). All GEMMs on the WMMA fp32 path.
// ---------------------------------------------------------------------------
extern "C" void kernel_launch(void* const* d_in, const int* in_sizes, int n_in,
                              void* d_out, int out_size, void* d_ws, size_t ws_size,
                              hipStream_t stream) {
    (void)in_sizes; (void)n_in; (void)out_size; (void)ws_size;

    const float* h1 = (const float*)d_in[0];   // (B,S,E)
    const float* h2 = (const float*)d_in[1];   // (B,S,E)
    const float* w  = (const float*)d_in[2];   // (2E,U)
    const float* b1 = (const float*)d_in[3];   // (U)
    const float* v  = (const float*)d_in[4];   // (U,1)
    const float* b2 = (const float*)d_in[5];   // (1)
    float* out = (float*)d_out;                // (B,S,E)

    float* qry_pre = (float*)d_ws;                        // B*S*U = 2 MB
    float* key_pre = qry_pre + (size_t)Bc * Sc * Uc;      // 2 MB
    float* P       = key_pre + (size_t)Bc * Sc * Uc;      // B*S*S = 4 MB

    dim3 blk(128);

    // key_pre = h1 (2048x512) @ w1 (512x256)
    wmma_gemm_f32<Uc, Ec, 0, 0, 0>
        <<<dim3(Uc / 64, (Bc * Sc) / 16, 1), blk, 0, stream>>>(h1, w, key_pre);

    // qry_pre = h2 (2048x512) @ w2 (512x256)
    wmma_gemm_f32<Uc, Ec, 0, 0, 0>
        <<<dim3(Uc / 64, (Bc * Sc) / 16, 1), blk, 0, stream>>>(
            h2, w + (size_t)Ec * Uc, qry_pre);

    // P = softmax_j( sum_u tanh(qry_pre + key_pre + b1) * v + b2 )
    attend_scores_softmax<<<Bc * Sc, 256, 0, stream>>>(
        qry_pre, key_pre, b1, v, b2, P);

    // out[b] = P[b] (512x512) @ h1[b] (512x512)
    wmma_gemm_f32<Ec, Sc, (long)Sc * Sc, (long)Sc * Ec, (long)Sc * Ec>
        <<<dim3(Ec / 64, Sc / 16, Bc), blk, 0, stream>>>(P, h1, out);
}